// SoftAttentionHHV_1202590843040
// MI455X (gfx1250) — compile-verified
//
#include <hip/hip_runtime.h>
#include <stdint.h>

#define NWIN 64
#define BLK  256
#define TILE (BLK + NWIN - 1)   // 319 floats staged per block

typedef unsigned int v4u __attribute__((ext_vector_type(4)));
typedef int          v4i __attribute__((ext_vector_type(4)));
typedef int          v8i __attribute__((ext_vector_type(8)));
typedef float        v2f __attribute__((ext_vector_type(2)));

static __device__ __forceinline__ float max3f(float a, float b, float c) {
    return fmaxf(fmaxf(a, b), c);   // v_max3 if the backend fuses, v_max otherwise
}

__global__ __launch_bounds__(BLK) void soft_attn_win64(
    const float* __restrict__ x,
    const float* __restrict__ wtau,
    float* __restrict__ out,
    int N)
{
    __shared__ float sm[TILE + 1];

    const int tid    = threadIdx.x;
    const int bstart = (int)blockIdx.x * BLK;          // N <= 2^31, int math only
    const int gstart = bstart - (NWIN - 1);            // first global idx of tile
    const int cstart = gstart < 0 ? 0 : gstart;        // clamped (left edge)
    int cend = bstart + BLK; if (cend > N) cend = N;   // clamped (right edge)
    const int nload = cend - cstart;                   // elements DMA'd
    const int shift = cstart - gstart;                 // 0..63 (block 0 only)

    // Finite sentinel for padded slots: exp2 arg overflows to -inf -> weight 0,
    // and fma(sentinel, 0, num) stays finite-zero, matching the reference mask.
    if (tid < shift) sm[tid] = -3.0e38f;

#if __has_builtin(__builtin_amdgcn_tensor_load_to_lds)
    // --- CDNA5 Tensor Data Mover: DMA the tile global -> LDS (one op/block) ---
    if (tid < 32) {  // single wave issues; TENSOR ops ignore EXEC
        const unsigned int       ldsb = (unsigned int)reinterpret_cast<uintptr_t>(&sm[shift]);
        const unsigned long long ga   = (unsigned long long)reinterpret_cast<uintptr_t>(x + cstart);
        const unsigned un = (unsigned)nload;

        v4u g0;  // D# group 0: control / lds_addr / global_addr / type
        g0[0] = 1u;                                             // count=1 (valid user D#)
        g0[1] = ldsb;                                           // lds_addr
        g0[2] = (unsigned)ga;                                   // global_addr lo
        g0[3] = ((unsigned)((ga >> 32) & 0x01FFFFFFull))        // global_addr [56:32]
                | 0x80000000u;                                  // type=2 ("image")

        v8i g1;  // D# group 1: sizes/strides, 1-D tile of `nload` f32 elements
        g1[0] = 0x00020000;                                     // data_size=2 (4 bytes)
        g1[1] = (int)((un & 0xFFFFu) << 16);                    // tensor_dim0[15:0]
        g1[2] = (int)(((un >> 16) & 0xFFFFu) | (1u << 16));     // tensor_dim0[31:16], tensor_dim1=1
        g1[3] = (int)((un & 0xFFFFu) << 16);                    // tile_dim0 = nload
        g1[4] = 0;                                              // tile_dim1=0, tile_dim2=0 (1-D)
        g1[5] = (int)un;                                        // tensor_dim0_stride lo
        g1[6] = 0;
        g1[7] = 0;

        v4i gz; gz[0] = gz[1] = gz[2] = gz[3] = 0;              // groups 2/3 unused (<=2D)
#if defined(__clang_major__) && (__clang_major__ >= 23)
        v8i gz8; gz8[0]=gz8[1]=gz8[2]=gz8[3]=gz8[4]=gz8[5]=gz8[6]=gz8[7]=0;
        __builtin_amdgcn_tensor_load_to_lds(g0, g1, gz, gz, gz8, 0);
#else
        __builtin_amdgcn_tensor_load_to_lds(g0, g1, gz, gz, 0);
#endif
        __builtin_amdgcn_s_wait_tensorcnt(0);
    }
#else
    // Fallback: cooperative copy
    for (int j = tid; j < nload; j += BLK) sm[shift + j] = x[cstart + j];
#endif
    __syncthreads();

    const int i = bstart + tid;
    if (i < N) {
        // tau = softplus(w_tau) + 1e-5 ; fold 1/(tau*ln2) into one exp2 scale
        const float tau = log1pf(expf(*wtau)) + 1e-5f;
        const float c   = 1.44269504088896340736f / tau;

        // Window lives in VGPRs between the two passes.
        float w[NWIN];
        #pragma unroll
        for (int k = 0; k < NWIN; ++k) w[k] = sm[tid + k];

        // 64-way max tree (v_max3 triples where the backend fuses them).
        float r[22];
        #pragma unroll
        for (int k = 0; k < 21; ++k) r[k] = max3f(w[3*k], w[3*k+1], w[3*k+2]);
        r[21] = w[63];
        float s[8];
        #pragma unroll
        for (int k = 0; k < 7; ++k) s[k] = max3f(r[3*k], r[3*k+1], r[3*k+2]);
        s[7] = r[21];
        const float m = max3f(max3f(s[0], s[1], s[2]),
                              max3f(s[3], s[4], s[5]),
                              fmaxf(s[6], s[7]));

        // exp pass on packed f32 pairs: v_pk_fma_f32 / v_pk_add_f32 halve the
        // non-transcendental VALU work; the two v_exp_f32 per pair stay scalar
        // (no packed trans unit).  (w - m)*c == fma(w, c, -m*c); the rounding
        // of m*c is a common shift across all slots and cancels in num/den.
        const float mc = m * c;
        const v2f c2   = { c, c };
        const v2f nmc2 = { -mc, -mc };
        v2f num2 = { 0.0f, 0.0f };
        v2f den2 = { 0.0f, 0.0f };
        #pragma unroll
        for (int j = 0; j < NWIN / 2; ++j) {
            v2f wv; wv.x = w[2*j]; wv.y = w[2*j + 1];
            const v2f arg = __builtin_elementwise_fma(wv, c2, nmc2);  // v_pk_fma_f32
            v2f ev;
            ev.x = __builtin_amdgcn_exp2f(arg.x);                     // v_exp_f32
            ev.y = __builtin_amdgcn_exp2f(arg.y);                     // v_exp_f32
            den2 += ev;                                               // v_pk_add_f32
            num2 = __builtin_elementwise_fma(wv, ev, num2);           // v_pk_fma_f32
        }
        const float num = num2.x + num2.y;
        const float den = den2.x + den2.y;
        out[i] = num / den;   // den >= 1 (the max element contributes e ~= 1)
    }
}

extern "C" void kernel_launch(void* const* d_in, const int* in_sizes, int n_in,
                              void* d_out, int out_size, void* d_ws, size_t ws_size,
                              hipStream_t stream) {
    const float* x    = (const float*)d_in[0];
    const float* wtau = (const float*)d_in[1];
    float* out        = (float*)d_out;
    const int N = in_sizes[0];
    const int nblk = (N + BLK - 1) / BLK;
    soft_attn_win64<<<dim3(nblk), dim3(BLK), 0, stream>>>(x, wtau, out, N);
}